// HTGNN_3006477107336
// MI455X (gfx1250) — compile-verified
//
#include <hip/hip_runtime.h>
#include <cstring>

// ---------------- problem constants ----------------
constexpr int Nn   = 50000;
constexpr int Ee   = 300000;
constexpr int Kk   = 3;
constexpr int INf  = 64;
constexpr int TDd  = 32;
constexpr int HIDd = 128;
constexpr int Hh   = 8;
constexpr int Dd   = 16;   // HID / H
#define CTf 8.0e5f

typedef float v2f __attribute__((ext_vector_type(2)));
typedef float v8f __attribute__((ext_vector_type(8)));

// monotonic float <-> uint encoding for atomic segment-max over signed floats
__device__ __forceinline__ unsigned fenc(float f) {
  unsigned b = __float_as_uint(f);
  return (b & 0x80000000u) ? ~b : (b | 0x80000000u);
}
__device__ __forceinline__ float fdec(unsigned k) {
  return __uint_as_float((k & 0x80000000u) ? (k & 0x7fffffffu) : ~k);
}

// ---------------- utility fills ----------------
__global__ void fill_f32(float* __restrict__ p, float v, int n) {
  int i = blockIdx.x * blockDim.x + threadIdx.x;
  if (i < n) p[i] = v;
}
__global__ void fill_u32(unsigned* __restrict__ p, unsigned v, int n) {
  int i = blockIdx.x * blockDim.x + threadIdx.x;
  if (i < n) p[i] = v;
}

// ---------------- temporal edge encoding ----------------
// per-edge: sinusoidal encode of dt, 32x32 projection (W in LDS), atomic segment sum
__global__ void __launch_bounds__(256)
time_edge(const float* __restrict__ ts, const int* __restrict__ dst,
          const float* __restrict__ W, const float* __restrict__ bvec,
          float* __restrict__ tsum, float* __restrict__ tcnt) {
  __shared__ float sW[TDd * TDd];
  __shared__ float sb[TDd];
  for (int i = threadIdx.x; i < TDd * TDd; i += blockDim.x) sW[i] = W[i];
  if (threadIdx.x < TDd) sb[threadIdx.x] = bvec[threadIdx.x];
  __syncthreads();
  int e = blockIdx.x * blockDim.x + threadIdx.x;
  if (e >= Ee) return;
  float t = ts[e];
  if (t > CTf) return;                       // masked edge: contributes nothing
  float dtv = (CTf - t) * (1.0f / 86400.0f);
  float raw[TDd];
  #pragma unroll
  for (int i = 0; i < 16; ++i) {
    // inv_freq = 10000^(-i/16) ; ln(10000)/16 = 0.5756462732485114
    float invf = __expf(-0.5756462732485114f * (float)i);
    float ang = dtv * invf;
    raw[i]      = __sinf(ang);
    raw[i + 16] = __cosf(ang);
  }
  int d = dst[e];
  float* srow = tsum + (size_t)d * TDd;
  #pragma unroll 4
  for (int j = 0; j < TDd; ++j) {
    float acc = sb[j];
    #pragma unroll
    for (int i = 0; i < TDd; ++i) acc += raw[i] * sW[i * TDd + j];
    atomicAdd(&srow[j], acc);
  }
  atomicAdd(&tcnt[d], 1.0f);
}

__global__ void __launch_bounds__(256)
build_xcomb(const float* __restrict__ feat, const float* __restrict__ tsum,
            const float* __restrict__ tcnt, float* __restrict__ xc) {
  int n = blockIdx.x * blockDim.x + threadIdx.x;
  if (n >= Nn) return;
  float* xr = xc + (size_t)n * (INf + TDd);
  const float* fr = feat + (size_t)n * INf;
  for (int j = 0; j < INf; ++j) xr[j] = fr[j];
  for (int j = 0; j < TDd; ++j) {
    float s = 0.f;
    for (int k = 0; k < Kk; ++k)
      s += tsum[((size_t)k * Nn + n) * TDd + j] / fmaxf(tcnt[(size_t)k * Nn + n], 1.0f);
    xr[INf + j] = s * (1.0f / Kk);
  }
}

// ---------------- WMMA f32 GEMM:  C[M x 128] = A[M x Fin] * B[Fin x 128] ----------------
// one wave -> one 16x16 tile; block = 8 waves covering the 8 column tiles of one row strip
__global__ void __launch_bounds__(256)
wmma_gemm_f32(const float* __restrict__ A, const float* __restrict__ B,
              float* __restrict__ C, const float* __restrict__ bias,
              int Fin, int relu) {
  const int wave = threadIdx.x >> 5;
  const int lane = threadIdx.x & 31;
  const int m0 = blockIdx.x << 4;
  const int n0 = wave << 4;
  const int r15 = lane & 15;
  const int kh  = (lane >> 4) << 1;      // 0 for lanes 0-15, 2 for lanes 16-31
  v8f c = {};
  const float* Ap = A + (size_t)(m0 + r15) * Fin + kh;
  const float* Bp = B + (size_t)kh * HIDd + n0 + r15;
  for (int k0 = 0; k0 < Fin; k0 += 4) {
    v2f a, b;
    a.x = Ap[0];  a.y = Ap[1];           // A[row][k0+kh], A[row][k0+kh+1]
    b.x = Bp[0];  b.y = Bp[HIDd];        // B[k0+kh][col], B[k0+kh+1][col]
#if defined(__HIP_DEVICE_COMPILE__)
    c = __builtin_amdgcn_wmma_f32_16x16x4_f32(false, a, false, b,
                                              (short)0, c, false, false);
#endif
    Ap += 4; Bp += 4 * HIDd;
  }
  const int rbase = (lane >> 4) << 3;    // rows 0..7 or 8..15
  float badd = bias ? bias[n0 + r15] : 0.f;
  #pragma unroll
  for (int r = 0; r < 8; ++r) {
    float v = c[r] + badd;
    if (relu) v = fmaxf(v, 0.f);
    C[(size_t)(m0 + rbase + r) * HIDd + n0 + r15] = v;
  }
}

// ---------------- per-node attention logits ----------------
__global__ void __launch_bounds__(256)
attn_logits(const float* __restrict__ h, const float* __restrict__ al,
            const float* __restrict__ ar, float* __restrict__ el,
            float* __restrict__ er) {
  int id = blockIdx.x * blockDim.x + threadIdx.x;
  if (id >= Nn * Hh) return;
  int n = id >> 3, hh = id & 7;
  const float* hp  = h + (size_t)n * HIDd + hh * Dd;
  const float* alp = al + hh * Dd;
  const float* arp = ar + hh * Dd;
  float a = 0.f, b = 0.f;
  #pragma unroll
  for (int d = 0; d < Dd; ++d) { a += hp[d] * alp[d]; b += hp[d] * arp[d]; }
  el[id] = a; er[id] = b;
}

// ---------------- GAT edge passes ----------------
__global__ void __launch_bounds__(256)
gat_passA(const float* __restrict__ ts, const int* __restrict__ src,
          const int* __restrict__ dst, const float* __restrict__ el,
          const float* __restrict__ er, unsigned* __restrict__ mkey,
          float* __restrict__ pe) {
  int e = blockIdx.x * blockDim.x + threadIdx.x;
  if (e >= Ee) return;
  bool mask = ts[e] <= CTf;
  int s = src[e], d = dst[e];
  #pragma unroll
  for (int hh = 0; hh < Hh; ++hh) {
    float ev = el[s * Hh + hh] + er[d * Hh + hh];
    ev = (ev > 0.f) ? ev : 0.2f * ev;            // leaky_relu(0.2)
    if (!mask) ev = -1.0e30f;
    pe[(size_t)e * Hh + hh] = ev;
    if (mask) atomicMax(&mkey[d * Hh + hh], fenc(ev));
  }
}

__global__ void __launch_bounds__(256)
gat_passB(const int* __restrict__ dst, const unsigned* __restrict__ mkey,
          float* __restrict__ pe, float* __restrict__ den) {
  int e = blockIdx.x * blockDim.x + threadIdx.x;
  if (e >= Ee) return;
  int d = dst[e];
  #pragma unroll
  for (int hh = 0; hh < Hh; ++hh) {
    float m = fdec(mkey[d * Hh + hh]);
    float mm = (m < -1.0e29f) ? 0.f : m;
    float p = __expf(pe[(size_t)e * Hh + hh] - mm);  // masked (-1e30) underflows to 0
    pe[(size_t)e * Hh + hh] = p;
    if (p > 0.f) atomicAdd(&den[d * Hh + hh], p);
  }
}

__global__ void __launch_bounds__(256)
gat_passC(const int* __restrict__ src, const int* __restrict__ dst,
          const float* __restrict__ pe, const float* __restrict__ den,
          const float* __restrict__ h, float* __restrict__ agg) {
  int id = blockIdx.x * blockDim.x + threadIdx.x;   // (edge, head)
  if (id >= Ee * Hh) return;
  int e = id >> 3, hh = id & 7;
  float p = pe[id];
  if (p <= 0.f) return;
  int d = dst[e];
  float alpha = p / fmaxf(den[d * Hh + hh], 1e-9f);
  int s = src[e];
  const float* hp = h + (size_t)s * HIDd + hh * Dd;
  float* op = agg + (size_t)d * HIDd + hh * Dd;
  #pragma unroll
  for (int i = 0; i < Dd; ++i) atomicAdd(&op[i], alpha * hp[i]);
}

// ---------------- node epilogues ----------------
__global__ void __launch_bounds__(256)
epi_relu_bias(const float* __restrict__ agg, const float* __restrict__ bias,
              float* __restrict__ outp) {
  int id = blockIdx.x * blockDim.x + threadIdx.x;
  if (id >= Nn * HIDd) return;
  int c = id & (HIDd - 1);
  float b = bias[c] + bias[HIDd + c] + bias[2 * HIDd + c];   // sum over etypes
  outp[id] = fmaxf(agg[id] + b, 0.f);
}

// residual + LayerNorm; one wave per node, 4 channels per lane
__global__ void __launch_bounds__(256)
ln_kernel(const float* __restrict__ h2, const float* __restrict__ bias2,
          const float* __restrict__ h1, const float* __restrict__ lw,
          const float* __restrict__ lb, float* __restrict__ hn) {
  int lane = threadIdx.x & 31;
  int node = (blockIdx.x * blockDim.x + threadIdx.x) >> 5;
  if (node >= Nn) return;
  float v[4]; float s = 0.f, sq = 0.f;
  #pragma unroll
  for (int i = 0; i < 4; ++i) {
    int c = lane + 32 * i;
    float b = bias2[c] + bias2[HIDd + c] + bias2[2 * HIDd + c];
    float x = h2[(size_t)node * HIDd + c] + b + h1[(size_t)node * HIDd + c];
    v[i] = x; s += x; sq += x * x;
  }
  #pragma unroll
  for (int off = 16; off; off >>= 1) {
    s  += __shfl_xor(s,  off, 32);
    sq += __shfl_xor(sq, off, 32);
  }
  float mu  = s * (1.f / HIDd);
  float var = sq * (1.f / HIDd) - mu * mu;
  float rinv = rsqrtf(var + 1e-5f);
  #pragma unroll
  for (int i = 0; i < 4; ++i) {
    int c = lane + 32 * i;
    hn[(size_t)node * HIDd + c] = (v[i] - mu) * rinv * lw[c] + lb[c];
  }
}

__global__ void __launch_bounds__(256)
clf_final(const float* __restrict__ z, const float* __restrict__ W2,
          const float* __restrict__ b2, float* __restrict__ out) {
  int n = blockIdx.x * blockDim.x + threadIdx.x;
  if (n >= Nn) return;
  const float* zp = z + (size_t)n * HIDd;
  float acc = b2[0];
  #pragma unroll 8
  for (int j = 0; j < HIDd; ++j) acc += zp[j] * W2[j];
  out[n] = acc;
}

// ---------------- host launcher ----------------
extern "C" void kernel_launch(void* const* d_in, const int* in_sizes, int n_in,
                              void* d_out, int out_size, void* d_ws, size_t ws_size,
                              hipStream_t stream) {
  (void)in_sizes; (void)n_in; (void)out_size; (void)ws_size;
  const float* feat   = (const float*)d_in[0];
  const float* ts     = (const float*)d_in[1];
  const int*   src    = (const int*)  d_in[2];
  const int*   dst    = (const int*)  d_in[3];
  const float* eprojW = (const float*)d_in[4];
  const float* eprojb = (const float*)d_in[5];
  const float* fc1W   = (const float*)d_in[6];
  const float* a1l    = (const float*)d_in[7];
  const float* a1r    = (const float*)d_in[8];
  const float* bias1  = (const float*)d_in[9];
  const float* fc2W   = (const float*)d_in[10];
  const float* a2l    = (const float*)d_in[11];
  const float* a2r    = (const float*)d_in[12];
  const float* bias2  = (const float*)d_in[13];
  const float* lnw    = (const float*)d_in[14];
  const float* lnb    = (const float*)d_in[15];
  const float* cW1    = (const float*)d_in[16];
  const float* cb1    = (const float*)d_in[17];
  const float* cW2    = (const float*)d_in[18];
  const float* cb2    = (const float*)d_in[19];
  float* out = (float*)d_out;

  // workspace carve-up (floats)
  float* ws = (float*)d_ws;
  size_t off = 0;
  float* xcomb = ws + off; off += (size_t)Nn * 96;
  float* h     = ws + off; off += (size_t)Nn * 128;   // per-etype h; later reused for hn
  float* h1    = ws + off; off += (size_t)Nn * 128;
  float* agg   = ws + off; off += (size_t)Nn * 128;
  float* z     = ws + off; off += (size_t)Nn * 128;
  float* el    = ws + off; off += (size_t)Nn * Hh;
  float* er    = ws + off; off += (size_t)Nn * Hh;
  unsigned* mkey = (unsigned*)(ws + off); off += (size_t)Nn * Hh;
  float* den   = ws + off; off += (size_t)Nn * Hh;
  float* pe    = ws + off; off += (size_t)Ee * Hh;
  float* tsum  = ws + off; off += (size_t)Kk * Nn * TDd;
  float* tcnt  = ws + off; off += (size_t)Kk * Nn;

  const int TB = 256;
  auto cdiv = [](long a, long b) { return (int)((a + b - 1) / b); };

  // encoded value of -1e30 for segment-max init (negative -> ~bits)
  float negbig = -1.0e30f;
  unsigned nb; std::memcpy(&nb, &negbig, 4);
  const unsigned ENC_NEG = ~nb;

  // ---- 1) temporal aggregation ----
  fill_f32<<<cdiv((long)Kk * Nn * TDd, TB), TB, 0, stream>>>(tsum, 0.f, Kk * Nn * TDd);
  fill_f32<<<cdiv((long)Kk * Nn, TB), TB, 0, stream>>>(tcnt, 0.f, Kk * Nn);
  for (int k = 0; k < Kk; ++k)
    time_edge<<<cdiv(Ee, TB), TB, 0, stream>>>(
        ts + (size_t)k * Ee, dst + (size_t)k * Ee,
        eprojW + (size_t)k * TDd * TDd, eprojb + (size_t)k * TDd,
        tsum + (size_t)k * Nn * TDd, tcnt + (size_t)k * Nn);
  build_xcomb<<<cdiv(Nn, TB), TB, 0, stream>>>(feat, tsum, tcnt, xcomb);

  // ---- 2) GAT layer 1 (input xcomb, Fin=96) ----
  fill_f32<<<cdiv((long)Nn * 128, TB), TB, 0, stream>>>(agg, 0.f, Nn * 128);
  for (int k = 0; k < Kk; ++k) {
    wmma_gemm_f32<<<Nn / 16, 256, 0, stream>>>(xcomb, fc1W + (size_t)k * 96 * 128, h,
                                               nullptr, 96, 0);
    attn_logits<<<cdiv((long)Nn * Hh, TB), TB, 0, stream>>>(h, a1l + k * 128, a1r + k * 128, el, er);
    fill_u32<<<cdiv((long)Nn * Hh, TB), TB, 0, stream>>>(mkey, ENC_NEG, Nn * Hh);
    fill_f32<<<cdiv((long)Nn * Hh, TB), TB, 0, stream>>>(den, 0.f, Nn * Hh);
    gat_passA<<<cdiv(Ee, TB), TB, 0, stream>>>(ts + (size_t)k * Ee, src + (size_t)k * Ee,
                                               dst + (size_t)k * Ee, el, er, mkey, pe);
    gat_passB<<<cdiv(Ee, TB), TB, 0, stream>>>(dst + (size_t)k * Ee, mkey, pe, den);
    gat_passC<<<cdiv((long)Ee * Hh, TB), TB, 0, stream>>>(src + (size_t)k * Ee,
                                                          dst + (size_t)k * Ee, pe, den, h, agg);
  }
  epi_relu_bias<<<cdiv((long)Nn * 128, TB), TB, 0, stream>>>(agg, bias1, h1);

  // ---- 3) GAT layer 2 (input h1, Fin=128) ----
  fill_f32<<<cdiv((long)Nn * 128, TB), TB, 0, stream>>>(agg, 0.f, Nn * 128);
  for (int k = 0; k < Kk; ++k) {
    wmma_gemm_f32<<<Nn / 16, 256, 0, stream>>>(h1, fc2W + (size_t)k * 128 * 128, h,
                                               nullptr, 128, 0);
    attn_logits<<<cdiv((long)Nn * Hh, TB), TB, 0, stream>>>(h, a2l + k * 128, a2r + k * 128, el, er);
    fill_u32<<<cdiv((long)Nn * Hh, TB), TB, 0, stream>>>(mkey, ENC_NEG, Nn * Hh);
    fill_f32<<<cdiv((long)Nn * Hh, TB), TB, 0, stream>>>(den, 0.f, Nn * Hh);
    gat_passA<<<cdiv(Ee, TB), TB, 0, stream>>>(ts + (size_t)k * Ee, src + (size_t)k * Ee,
                                               dst + (size_t)k * Ee, el, er, mkey, pe);
    gat_passB<<<cdiv(Ee, TB), TB, 0, stream>>>(dst + (size_t)k * Ee, mkey, pe, den);
    gat_passC<<<cdiv((long)Ee * Hh, TB), TB, 0, stream>>>(src + (size_t)k * Ee,
                                                          dst + (size_t)k * Ee, pe, den, h, agg);
  }

  // ---- 4) residual + LayerNorm (hn stored in h, free now) ----
  ln_kernel<<<cdiv((long)Nn * 32, TB), TB, 0, stream>>>(agg, bias2, h1, lnw, lnb, h);

  // ---- 5) classifier: z = relu(hn @ W1 + b1); out = z @ W2 + b2 ----
  wmma_gemm_f32<<<Nn / 16, 256, 0, stream>>>(h, cW1, z, cb1, 128, 1);
  clf_final<<<cdiv(Nn, TB), TB, 0, stream>>>(z, cW2, cb2, out);
}